// SNNLifRefrac_45792941310313
// MI455X (gfx1250) — compile-verified
//
#include <hip/hip_runtime.h>
#include <cstddef>

typedef float v2f __attribute__((ext_vector_type(2)));
typedef float v8f __attribute__((ext_vector_type(8)));

#define T_STEPS 1000
#define B_DIM   64
#define F_DIM   128
#define H_DIM   512
#define O_DIM   32

#define NBLOCKS  32
#define NTHREADS 128   // 4 wave32 per block -> 128 waves = 128 state tiles

#define NK_IN   (F_DIM / 4)   // 32 k-steps of K=4
#define NK_REC  (H_DIM / 4)   // 128 k-steps
#define NK_OUT  (H_DIM / 4)   // 128 k-steps

// ---------------------------------------------------------------------------
// fp32 WMMA: D(16x16) = A(16x4) * B(4x16) + C, full fp32 precision
// ---------------------------------------------------------------------------
__device__ __forceinline__ v8f wmma4_f32(v2f a, v2f b, v8f c) {
  return __builtin_amdgcn_wmma_f32_16x16x4_f32(false, a, false, b, (short)0, c,
                                               false, false);
}

// ---------------------------------------------------------------------------
// Device-wide sense-reversing barrier (persistent grid of NBLOCKS blocks).
// ---------------------------------------------------------------------------
__device__ __forceinline__ void grid_sync(unsigned* cnt, unsigned* gen) {
  __threadfence();     // publish our global stores device-wide
  __syncthreads();
  if (threadIdx.x == 0) {
    unsigned g0 = __hip_atomic_load(gen, __ATOMIC_RELAXED, __HIP_MEMORY_SCOPE_AGENT);
    unsigned prev = __hip_atomic_fetch_add(cnt, 1u, __ATOMIC_ACQ_REL,
                                           __HIP_MEMORY_SCOPE_AGENT);
    if (prev == NBLOCKS - 1) {
      __hip_atomic_store(cnt, 0u, __ATOMIC_RELAXED, __HIP_MEMORY_SCOPE_AGENT);
      __hip_atomic_fetch_add(gen, 1u, __ATOMIC_ACQ_REL, __HIP_MEMORY_SCOPE_AGENT);
    } else {
      while (__hip_atomic_load(gen, __ATOMIC_ACQUIRE,
                               __HIP_MEMORY_SCOPE_AGENT) == g0) {
        __builtin_amdgcn_s_sleep(1);
      }
    }
  }
  __syncthreads();
  __threadfence();     // acquire: drop stale cached z lines before re-reading
}

// ---------------------------------------------------------------------------
// Persistent scan kernel. Wave W owns state tile (b0 = 16*(W&3), h0 = 16*(W>>2)).
// h_tile == blockIdx.x for all 4 waves -> weight B-fragments are block-shared:
// staged once into LDS in fragment-packed order [kstep][lane] (conflict-free
// ds_load_b64 in the hot loop). z is triple-buffered in global; one grid
// barrier per timestep.
// ---------------------------------------------------------------------------
__global__ void __launch_bounds__(NTHREADS)
snn_lif_refrac_kernel(const float* __restrict__ x,      // [T,B,F]
                      const float* __restrict__ w_in,   // [H,F]
                      const float* __restrict__ w_rec,  // [H,H]
                      const float* __restrict__ w_out,  // [O,H]
                      const float* __restrict__ b_out,  // [O]
                      float* __restrict__ out,          // [T,B,O]
                      float* __restrict__ zbuf,         // [3][B][H]
                      unsigned* __restrict__ bar_cnt,
                      unsigned* __restrict__ bar_gen) {
  __shared__ float lds_win [NK_IN  * 64];   //  8 KB: w_in  fragments
  __shared__ float lds_wrec[NK_REC * 64];   // 32 KB: w_rec fragments
  __shared__ float lds_wout[NK_OUT * 64];   // 32 KB: w_out fragments (blk 0,1)

  const int lane = threadIdx.x & 31;
  const int wave = threadIdx.x >> 5;
  const int W    = blockIdx.x * (NTHREADS / 32) + wave;  // 0..127
  const int l15  = lane & 15;
  const int hi   = lane >> 4;                            // 0 or 1

  const int b0 = (W & 3) * 16;   // batch-row block of this wave's tile
  const int h0 = blockIdx.x * 16;  // hidden-col block (uniform per block)

  const bool is_ro = (W < 8);        // 8 readout tiles: 4 b-blocks x 2 o-blocks
  const int  ro_n0 = blockIdx.x * 16;  // 0 or 16 for blocks 0,1

  // ---- stage time-invariant B-fragments into LDS (fragment-packed) --------
  {
    v2f* fw = (v2f*)lds_win;
    for (int kk = wave; kk < NK_IN; kk += 4)
      fw[kk * 32 + lane] =
          *(const v2f*)(w_in + (size_t)(h0 + l15) * F_DIM + kk * 4 + 2 * hi);
    v2f* fr = (v2f*)lds_wrec;
    for (int kk = wave; kk < NK_REC; kk += 4)
      fr[kk * 32 + lane] =
          *(const v2f*)(w_rec + (size_t)(h0 + l15) * H_DIM + kk * 4 + 2 * hi);
    if (blockIdx.x < 2) {
      v2f* fo = (v2f*)lds_wout;
      for (int kk = wave; kk < NK_OUT; kk += 4)
        fo[kk * 32 + lane] =
            *(const v2f*)(w_out + (size_t)(ro_n0 + l15) * H_DIM + kk * 4 + 2 * hi);
    }
  }
  __syncthreads();

  const v2f* wi = ((const v2f*)lds_win)  + lane;   // stride 32 v2f per kstep
  const v2f* wr = ((const v2f*)lds_wrec) + lane;
  const v2f* wo = ((const v2f*)lds_wout) + lane;

  // state in WMMA C-fragment layout: element r -> (row b0+r+8*hi, col h0+l15)
  float v_s[8], i_s[8], rho_s[8], o_s[8];
#pragma unroll
  for (int r = 0; r < 8; ++r) { v_s[r] = 0.f; i_s[r] = 0.f; rho_s[r] = 0.f; o_s[r] = 0.f; }

  const float bout_l = is_ro ? b_out[ro_n0 + l15] : 0.f;

  constexpr float KMEM = 0.1f;             // DT * TAU_MEM_INV
  constexpr float KSYN = 0.8f;             // 1 - DT * TAU_SYN_INV
  constexpr float KFIL = 0.2231435511314f; // DT * TAU_FILTER_INV
  constexpr int   BH   = B_DIM * H_DIM;

  int cur = 0, prev = 2;  // rotating z-buffer indices (t%3, (t-1)%3)

#pragma unroll 1
  for (int t = 0; t < T_STEPS; ++t) {
    float* zw = zbuf + (size_t)cur * BH;

    // ---- phase A: elementwise LIF + refractory, publish z(t) --------------
#pragma unroll
    for (int r = 0; r < 8; ++r) {
      float v = v_s[r], curI = i_s[r], rho = rho_s[r];
      float v_dec = v + KMEM * (curI - v);
      float i_dec = curI * KSYN;
      float z_pre = (v_dec > 1.0f) ? 1.0f : 0.0f;   // heaviside(v_dec - V_TH)
      float mask  = (rho > 0.0f) ? 1.0f : 0.0f;     // heaviside(rho)
      float v_sp  = (1.0f - z_pre) * v_dec;         // V_RESET = 0
      v_s[r]      = (1.0f - mask) * v_sp + mask * v;
      float z_new = (1.0f - mask) * z_pre;
      rho_s[r]    = (1.0f - z_new) * fmaxf(rho - mask, 0.0f) + z_new * 5.0f;
      i_s[r]      = i_dec;                          // GEMM added in phase C
      zw[(size_t)(b0 + r + 8 * hi) * H_DIM + (h0 + l15)] = z_new;
    }

    // prefetch next timestep's x slab; latency hides behind the barrier
    if (t + 1 < T_STEPS) {
      const char* nx =
          (const char*)(x + ((size_t)(t + 1) * B_DIM + (b0 + l15)) * F_DIM) +
          hi * 256;
      __builtin_prefetch(nx, 0, 1);
    }

    grid_sync(bar_cnt, bar_gen);   // single barrier/step (triple-buffered z)

    // ---- phase C: i(t) = i_dec + x_t @ w_in^T + z(t-1) @ w_rec^T ----------
    v8f acc0 = {0.f, 0.f, 0.f, 0.f, 0.f, 0.f, 0.f, 0.f};
    v8f acc1 = acc0;

    const float* xa = x + ((size_t)t * B_DIM + (b0 + l15)) * F_DIM + 2 * hi;
#pragma unroll 4
    for (int kk = 0; kk < NK_IN; kk += 2) {
      v2f a0 = *(const v2f*)(xa + 4 * kk);
      acc0 = wmma4_f32(a0, wi[kk * 32], acc0);
      v2f a1 = *(const v2f*)(xa + 4 * kk + 4);
      acc1 = wmma4_f32(a1, wi[(kk + 1) * 32], acc1);
    }

    if (t > 0) {  // wave-uniform branch; z(-1) == 0
      const float* za =
          zbuf + (size_t)prev * BH + (size_t)(b0 + l15) * H_DIM + 2 * hi;
#pragma unroll 4
      for (int kk = 0; kk < NK_REC; kk += 2) {
        v2f a0 = *(const v2f*)(za + 4 * kk);
        acc0 = wmma4_f32(a0, wr[kk * 32], acc0);
        v2f a1 = *(const v2f*)(za + 4 * kk + 4);
        acc1 = wmma4_f32(a1, wr[(kk + 1) * 32], acc1);
      }
    }

#pragma unroll
    for (int r = 0; r < 8; ++r) i_s[r] += acc0[r] + acc1[r];

    // ---- readout: o += KFIL * ((z(t) @ w_out^T + b_out) - o) --------------
    if (is_ro) {  // wave-uniform branch (blocks 0,1)
      v8f r0 = {0.f, 0.f, 0.f, 0.f, 0.f, 0.f, 0.f, 0.f};
      v8f r1 = r0;
      const float* za = zw + (size_t)(b0 + l15) * H_DIM + 2 * hi;  // z(t)
#pragma unroll 4
      for (int kk = 0; kk < NK_OUT; kk += 2) {
        v2f a0 = *(const v2f*)(za + 4 * kk);
        r0 = wmma4_f32(a0, wo[kk * 32], r0);
        v2f a1 = *(const v2f*)(za + 4 * kk + 4);
        r1 = wmma4_f32(a1, wo[(kk + 1) * 32], r1);
      }
#pragma unroll
      for (int r = 0; r < 8; ++r) {
        float lin = r0[r] + r1[r] + bout_l;
        o_s[r] += KFIL * (lin - o_s[r]);
        out[((size_t)t * B_DIM + (b0 + r + 8 * hi)) * O_DIM + (ro_n0 + l15)] = o_s[r];
      }
    }

    prev = cur;
    cur  = (cur == 2) ? 0 : cur + 1;
  }
}

// ---------------------------------------------------------------------------
// Host launch. ws layout: [0,256) barrier words, [256, 256+3*B*H*4) z buffer.
// hipMemsetAsync is graph-capturable; only barrier words need zeroing.
// ---------------------------------------------------------------------------
extern "C" void kernel_launch(void* const* d_in, const int* in_sizes, int n_in,
                              void* d_out, int out_size, void* d_ws, size_t ws_size,
                              hipStream_t stream) {
  (void)in_sizes; (void)n_in; (void)out_size; (void)ws_size;
  const float* x     = (const float*)d_in[0];
  const float* w_in  = (const float*)d_in[1];
  const float* w_rec = (const float*)d_in[2];
  const float* w_out = (const float*)d_in[3];
  const float* b_out = (const float*)d_in[4];
  float* out = (float*)d_out;

  unsigned* bar  = (unsigned*)d_ws;
  float*    zbuf = (float*)((char*)d_ws + 256);

  hipMemsetAsync(d_ws, 0, 256, stream);

  hipLaunchKernelGGL(snn_lif_refrac_kernel, dim3(NBLOCKS), dim3(NTHREADS), 0, stream,
                     x, w_in, w_rec, w_out, b_out, out, zbuf, bar, bar + 1);
}